// PCTransformer1_22127671509192
// MI455X (gfx1250) — compile-verified
//
#include <hip/hip_runtime.h>

#define LEAK   0.2f
#define GN_EPS 1e-5f
#define KN     16
#define GROUPS 4

typedef __attribute__((ext_vector_type(2))) float v2f;
typedef __attribute__((ext_vector_type(8))) float v8f;

// ---------------------------------------------------------------------------
// Stage 0: coor = x^T (B,3,N) and f0 = w_in * coor + b_in (B,8,N)
// ---------------------------------------------------------------------------
__global__ __launch_bounds__(256) void prep_kernel(
    const float* __restrict__ x, const float* __restrict__ w_in,
    const float* __restrict__ b_in, float* __restrict__ coor0,
    float* __restrict__ f0, int B, int N) {
  int t = blockIdx.x * blockDim.x + threadIdx.x;
  if (t >= B * N) return;
  int n = t % N, b = t / N;
  const float* xp = x + ((size_t)b * N + n) * 3;
  float cx = xp[0], cy = xp[1], cz = xp[2];
  size_t base = (size_t)b * 3 * N + n;
  coor0[base] = cx; coor0[base + N] = cy; coor0[base + 2 * (size_t)N] = cz;
#pragma unroll
  for (int o = 0; o < 8; ++o) {
    float v = w_in[o * 3 + 0] * cx + w_in[o * 3 + 1] * cy + w_in[o * 3 + 2] * cz + b_in[o];
    f0[((size_t)b * 8 + o) * N + n] = v;
  }
}

// ---------------------------------------------------------------------------
// kNN (k=16): one thread per query, keys staged through LDS, register
// insertion sort (stable on ties -> matches lax.top_k ordering).
// ---------------------------------------------------------------------------
__global__ __launch_bounds__(128) void knn_kernel(
    const float* __restrict__ cq, const float* __restrict__ ck,
    int* __restrict__ out, int Nq, int Nk) {
  __shared__ float skx[128], sky[128], skz[128];
  int b = blockIdx.y;
  int q = blockIdx.x * blockDim.x + threadIdx.x;
  bool valid = q < Nq;
  const float* cqb = cq + (size_t)b * 3 * Nq;
  const float* ckb = ck + (size_t)b * 3 * Nk;
  float qx = 0.f, qy = 0.f, qz = 0.f;
  if (valid) { qx = cqb[q]; qy = cqb[Nq + q]; qz = cqb[2 * (size_t)Nq + q]; }
  float bd[KN]; int bi[KN];
#pragma unroll
  for (int t = 0; t < KN; ++t) { bd[t] = 3.4e38f; bi[t] = 0; }
  for (int base = 0; base < Nk; base += 128) {
    __syncthreads();
    int i = base + threadIdx.x;
    if (i < Nk) {
      skx[threadIdx.x] = ckb[i];
      sky[threadIdx.x] = ckb[Nk + i];
      skz[threadIdx.x] = ckb[2 * (size_t)Nk + i];
    }
    // prefetch next key tile into cache while we crunch this one
    int pn = base + 128 + threadIdx.x;
    if (pn < Nk) {
      __builtin_prefetch(&ckb[pn], 0, 0);
      __builtin_prefetch(&ckb[Nk + pn], 0, 0);
      __builtin_prefetch(&ckb[2 * (size_t)Nk + pn], 0, 0);
    }
    __syncthreads();
    int lim = min(128, Nk - base);
    if (valid) {
#pragma unroll 4
      for (int t = 0; t < lim; ++t) {
        float dx = qx - skx[t], dy = qy - sky[t], dz = qz - skz[t];
        float d = dx * dx + dy * dy + dz * dz;
        if (d < bd[KN - 1]) {
          bd[KN - 1] = d; bi[KN - 1] = base + t;
#pragma unroll
          for (int s = KN - 1; s > 0; --s) {
            if (bd[s] < bd[s - 1]) {
              float td = bd[s]; bd[s] = bd[s - 1]; bd[s - 1] = td;
              int ti = bi[s];   bi[s] = bi[s - 1]; bi[s - 1] = ti;
            }
          }
        }
      }
    }
  }
  if (valid) {
    int* o = out + ((size_t)b * Nq + q) * KN;
#pragma unroll
    for (int t = 0; t < KN; ++t) o[t] = bi[t];
  }
}

// ---------------------------------------------------------------------------
// Fused edge-feature gather + 1x1 conv via V_WMMA_F32_16X16X4_F32.
// One wave = one query point (16 neighbors = WMMA N dim) x 16 output channels.
// All geometry (CIN, OC, NQ, NK) is compile-time: the wave-index decomposition
// is pure shift/mask, channel strides fold into load immediates, and the
// K-loop unrolls so gather clauses overlap the WMMA chain.
// PASS 0: accumulate GroupNorm sum/sumsq.  PASS 1: normalize+lrelu+max_k+store.
// ---------------------------------------------------------------------------
template <int PASS, int CIN, int OC, int NQ, int NK>
__global__ __launch_bounds__(256) void edgeconv_wmma(
    const float* __restrict__ fq, const float* __restrict__ fk,
    const int* __restrict__ nidx, const float* __restrict__ W,
    const float* __restrict__ gamma, const float* __restrict__ beta,
    float* __restrict__ stats, const float* __restrict__ meanistd,
    float* __restrict__ fout, int B) {
  constexpr int C = 2 * CIN;          // edge-feature channels
  constexpr int OTILES = OC >> 4;
  const int wid  = (int)(threadIdx.x >> 5);
  const int lane = (int)(threadIdx.x & 31);
  const unsigned gw = blockIdx.x * (blockDim.x >> 5) + wid;  // fits 32-bit
  const unsigned total = (unsigned)B * NQ * OTILES;
  if (gw >= total) return;            // wave-uniform: EXEC stays all-ones
  const int ot = (int)(gw % OTILES);  // pow2 -> mask
  const unsigned t2 = gw / OTILES;    // pow2 -> shift
  const int n = (int)(t2 % NQ);       // pow2 -> mask
  const int b = (int)(t2 / NQ);       // pow2 -> shift
  const int o0 = ot << 4;
  const int hf  = lane >> 4;          // half-wave -> K sub-pair
  const int l15 = lane & 15;          // M (A) / neighbor j (B)
  const float* fqb = fq + (size_t)b * CIN * NQ + n;      // stride NQ per channel
  const int nbr = nidx[((size_t)b * NQ + n) * KN + l15];
  const float* fkn = fk + (size_t)b * CIN * NK + nbr;    // stride NK per channel
  const float* Wr  = W + (size_t)(o0 + l15) * C + 2 * hf;

  v8f acc = {0.f, 0.f, 0.f, 0.f, 0.f, 0.f, 0.f, 0.f};
#pragma unroll 8
  for (int c0 = 0; c0 < C; c0 += 4) {
    const int ca = c0 + 2 * hf;       // this lane's K pair: ca, ca+1
    v2f av, bv;
    av.x = Wr[c0];
    av.y = Wr[c0 + 1];
    float b0, b1;
    if (ca < CIN)     b0 = fkn[(size_t)ca * NK] - fqb[(size_t)ca * NQ];
    else              b0 = fqb[(size_t)(ca - CIN) * NQ];
    const int cb = ca + 1;
    if (cb < CIN)     b1 = fkn[(size_t)cb * NK] - fqb[(size_t)cb * NQ];
    else              b1 = fqb[(size_t)(cb - CIN) * NQ];
    bv.x = b0; bv.y = b1;
    acc = __builtin_amdgcn_wmma_f32_16x16x4_f32(false, av, false, bv,
                                                (short)0, acc, false, false);
  }

  constexpr int GSZ = OC / GROUPS;
  if (PASS == 0) {
    const int gbase = o0 / GSZ;
    const int gtop  = (o0 + 15) / GSZ;   // a 16-ch tile spans <=2 groups
    float s0 = 0.f, q0 = 0.f, s1 = 0.f, q1 = 0.f;
#pragma unroll
    for (int r = 0; r < 8; ++r) {
      int ch = o0 + r + 8 * hf;
      float v = acc[r];
      if ((ch / GSZ) == gbase) { s0 += v; q0 += v * v; }
      else                     { s1 += v; q1 += v * v; }
    }
#pragma unroll
    for (int off = 16; off > 0; off >>= 1) {
      s0 += __shfl_xor(s0, off, 32);
      q0 += __shfl_xor(q0, off, 32);
      s1 += __shfl_xor(s1, off, 32);
      q1 += __shfl_xor(q1, off, 32);
    }
    if (lane == 0) {
      atomicAdd(&stats[((size_t)b * GROUPS + gbase) * 2 + 0], s0);
      atomicAdd(&stats[((size_t)b * GROUPS + gbase) * 2 + 1], q0);
      if (gtop != gbase) {
        atomicAdd(&stats[((size_t)b * GROUPS + gtop) * 2 + 0], s1);
        atomicAdd(&stats[((size_t)b * GROUPS + gtop) * 2 + 1], q1);
      }
    }
  } else {
#pragma unroll
    for (int r = 0; r < 8; ++r) {
      int ch = o0 + r + 8 * hf;
      int g = ch / GSZ;
      float mu = meanistd[((size_t)b * GROUPS + g) * 2 + 0];
      float is = meanistd[((size_t)b * GROUPS + g) * 2 + 1];
      float v = (acc[r] - mu) * is * gamma[ch] + beta[ch];
      v = (v >= 0.f) ? v : LEAK * v;
#pragma unroll
      for (int off = 1; off < 16; off <<= 1)   // max over 16 neighbors (lanes)
        v = fmaxf(v, __shfl_xor(v, off, 32));
      if (l15 == 0) fout[((size_t)b * OC + ch) * NQ + n] = v;
    }
  }
}

// ---------------------------------------------------------------------------
// GroupNorm stat finalize: mean / inv_std per (b, group)
// ---------------------------------------------------------------------------
__global__ void gn_finalize(const float* __restrict__ stats,
                            float* __restrict__ meanistd,
                            float cnt_inv, int total) {
  int i = blockIdx.x * blockDim.x + threadIdx.x;
  if (i >= total) return;
  float s = stats[i * 2], q = stats[i * 2 + 1];
  float mu = s * cnt_inv;
  float var = q * cnt_inv - mu * mu;
  meanistd[i * 2] = mu;
  meanistd[i * 2 + 1] = rsqrtf(var + GN_EPS);
}

__global__ void zero_kernel(float* p, int n) {
  int i = blockIdx.x * blockDim.x + threadIdx.x;
  if (i < n) p[i] = 0.f;
}

// ---------------------------------------------------------------------------
// Farthest point sampling: one block per batch; dist kept in registers
// (<=8 points/thread), LDS tree argmax per iteration (ties -> lowest index).
// ---------------------------------------------------------------------------
__global__ __launch_bounds__(256) void fps_kernel(
    const float* __restrict__ coor, int N, int nsamp, int* __restrict__ out) {
  __shared__ float sv[256];
  __shared__ int   si[256];
  __shared__ float spt[3];
  int b = blockIdx.x, t = threadIdx.x;
  const float* cb = coor + (size_t)b * 3 * N;
  float px[8], py[8], pz[8], dd[8];
  float x0 = cb[0], y0 = cb[N], z0 = cb[2 * (size_t)N];
#pragma unroll
  for (int q = 0; q < 8; ++q) {
    int i = t + q * 256;
    if (i < N) {
      px[q] = cb[i]; py[q] = cb[N + i]; pz[q] = cb[2 * (size_t)N + i];
      float dx = px[q] - x0, dy = py[q] - y0, dz = pz[q] - z0;
      dd[q] = dx * dx + dy * dy + dz * dz;
    } else { px[q] = py[q] = pz[q] = 0.f; dd[q] = -1.f; }
  }
  int* ob = out + (size_t)b * nsamp;
  if (t == 0) ob[0] = 0;
  for (int it = 1; it < nsamp; ++it) {
    float bv = -1.f; int bix = 0x7fffffff;
#pragma unroll
    for (int q = 0; q < 8; ++q) {
      int i = t + q * 256;
      if (dd[q] > bv || (dd[q] == bv && i < bix)) { bv = dd[q]; bix = i; }
    }
    sv[t] = bv; si[t] = bix;
    __syncthreads();
    for (int off = 128; off > 0; off >>= 1) {
      if (t < off) {
        float ov = sv[t + off]; int oi = si[t + off];
        if (ov > sv[t] || (ov == sv[t] && oi < si[t])) { sv[t] = ov; si[t] = oi; }
      }
      __syncthreads();
    }
    int sel = si[0];
    if (t == 0) {
      ob[it] = sel;
      spt[0] = cb[sel]; spt[1] = cb[N + sel]; spt[2] = cb[2 * (size_t)N + sel];
    }
    __syncthreads();
    float sx = spt[0], sy = spt[1], sz = spt[2];
#pragma unroll
    for (int q = 0; q < 8; ++q) {
      int i = t + q * 256;
      if (i < N) {
        float dx = px[q] - sx, dy = py[q] - sy, dz = pz[q] - sz;
        dd[q] = fminf(dd[q], dx * dx + dy * dy + dz * dz);
      }
    }
    __syncthreads();
  }
}

// gather [coor;f] rows at FPS indices
__global__ __launch_bounds__(256) void gather_fps(
    const float* __restrict__ ci, const float* __restrict__ fi,
    const int* __restrict__ fidx, float* __restrict__ co, float* __restrict__ fo,
    int Nin, int Nout, int C, int B) {
  int tot = B * (3 + C) * Nout;
  int t = blockIdx.x * blockDim.x + threadIdx.x;
  if (t >= tot) return;
  int i = t % Nout;
  int c = (t / Nout) % (3 + C);
  int b = t / (Nout * (3 + C));
  int src = fidx[(size_t)b * Nout + i];
  if (c < 3) co[((size_t)b * 3 + c) * Nout + i] = ci[((size_t)b * 3 + c) * Nin + src];
  else fo[((size_t)b * C + (c - 3)) * Nout + i] = fi[((size_t)b * C + (c - 3)) * Nin + src];
}

// final transposed outputs: [B,N,3] coords then [B,N,C] features
__global__ __launch_bounds__(256) void write_out(
    const float* __restrict__ coor, const float* __restrict__ f,
    float* __restrict__ out, int B, int N, int C) {
  int tot = B * N * (3 + C);
  int t = blockIdx.x * blockDim.x + threadIdx.x;
  if (t >= tot) return;
  int c = t % (3 + C);
  int n = (t / (3 + C)) % N;
  int b = t / ((3 + C) * N);
  if (c < 3)
    out[((size_t)b * N + n) * 3 + c] = coor[((size_t)b * 3 + c) * N + n];
  else
    out[(size_t)B * N * 3 + ((size_t)b * N + n) * C + (c - 3)] =
        f[((size_t)b * C + (c - 3)) * N + n];
}

// ---------------------------------------------------------------------------
template <int CIN, int OC, int NQ, int NK>
static void run_edge(const float* fq, const float* fk, const int* idx,
                     const float* W, const float* gm, const float* bt,
                     float* stats, float* meanistd, float* fo,
                     int B, hipStream_t stream) {
  long waves = (long)B * NQ * (OC / 16);
  int blocks = (int)((waves + 7) / 8);
  zero_kernel<<<1, 256, 0, stream>>>(stats, B * GROUPS * 2);
  edgeconv_wmma<0, CIN, OC, NQ, NK><<<blocks, 256, 0, stream>>>(
      fq, fk, idx, W, gm, bt, stats, meanistd, fo, B);
  float cnt_inv = 1.0f / ((float)(OC / GROUPS) * (float)NQ * (float)KN);
  gn_finalize<<<1, 256, 0, stream>>>(stats, meanistd, cnt_inv, B * GROUPS);
  edgeconv_wmma<1, CIN, OC, NQ, NK><<<blocks, 256, 0, stream>>>(
      fq, fk, idx, W, gm, bt, stats, meanistd, fo, B);
}

extern "C" void kernel_launch(void* const* d_in, const int* in_sizes, int n_in,
                              void* d_out, int out_size, void* d_ws, size_t ws_size,
                              hipStream_t stream) {
  (void)in_sizes; (void)n_in; (void)out_size; (void)ws_size;
  const int B = 32, N0 = 2048, N1 = 512, N2 = 128;

  const float* x    = (const float*)d_in[0];
  const float* w_in = (const float*)d_in[1];
  const float* b_in = (const float*)d_in[2];
  const float* w1 = (const float*)d_in[3];
  const float* g1 = (const float*)d_in[4];
  const float* be1 = (const float*)d_in[5];
  const float* w2 = (const float*)d_in[6];
  const float* g2 = (const float*)d_in[7];
  const float* be2 = (const float*)d_in[8];
  const float* w3 = (const float*)d_in[9];
  const float* g3 = (const float*)d_in[10];
  const float* be3 = (const float*)d_in[11];
  const float* w4 = (const float*)d_in[12];
  const float* g4 = (const float*)d_in[13];
  const float* be4 = (const float*)d_in[14];
  float* out = (float*)d_out;

  // workspace layout (4-byte units)
  float* wsf = (float*)d_ws;
  size_t off = 0;
  auto A = [&](size_t n) { float* p = wsf + off; off += n; return p; };
  float* coor0 = A((size_t)B * 3 * N0);
  float* f0    = A((size_t)B * 8 * N0);
  float* f1    = A((size_t)B * 32 * N0);
  float* coorA = A((size_t)B * 3 * N1);
  float* fqA   = A((size_t)B * 32 * N1);
  float* f2    = A((size_t)B * 64 * N1);
  float* f3    = A((size_t)B * 64 * N1);
  float* coorB = A((size_t)B * 3 * N2);
  float* fqB   = A((size_t)B * 64 * N2);
  float* f4    = A((size_t)B * 256 * N2);
  int* idx1 = (int*)A((size_t)B * N0 * KN);
  int* idx2 = (int*)A((size_t)B * N1 * KN);
  int* idx3 = (int*)A((size_t)B * N1 * KN);
  int* idx4 = (int*)A((size_t)B * N2 * KN);
  int* fps0 = (int*)A((size_t)B * N1);
  int* fps1 = (int*)A((size_t)B * N2);
  float* stats    = A((size_t)B * GROUPS * 2);
  float* meanistd = A((size_t)B * GROUPS * 2);

  // stage 0: transpose + input projection
  prep_kernel<<<(B * N0 + 255) / 256, 256, 0, stream>>>(x, w_in, b_in, coor0, f0, B, N0);

  // stage 1: self kNN on 2048, edge-conv 16->32
  knn_kernel<<<dim3((N0 + 127) / 128, B), 128, 0, stream>>>(coor0, coor0, idx1, N0, N0);
  run_edge<8, 32, 2048, 2048>(f0, f0, idx1, w1, g1, be1, stats, meanistd, f1, B, stream);

  // FPS 2048 -> 512, gather
  fps_kernel<<<B, 256, 0, stream>>>(coor0, N0, N1, fps0);
  gather_fps<<<(B * 35 * N1 + 255) / 256, 256, 0, stream>>>(coor0, f1, fps0, coorA, fqA,
                                                            N0, N1, 32, B);

  // stage 2: 512 queries vs 2048 keys, edge-conv 64->64
  knn_kernel<<<dim3((N1 + 127) / 128, B), 128, 0, stream>>>(coorA, coor0, idx2, N1, N0);
  run_edge<32, 64, 512, 2048>(fqA, f1, idx2, w2, g2, be2, stats, meanistd, f2, B, stream);

  // stage 3: self kNN on 512, edge-conv 128->64
  knn_kernel<<<dim3((N1 + 127) / 128, B), 128, 0, stream>>>(coorA, coorA, idx3, N1, N1);
  run_edge<64, 64, 512, 512>(f2, f2, idx3, w3, g3, be3, stats, meanistd, f3, B, stream);

  // FPS 512 -> 128, gather
  fps_kernel<<<B, 256, 0, stream>>>(coorA, N1, N2, fps1);
  gather_fps<<<(B * 67 * N2 + 255) / 256, 256, 0, stream>>>(coorA, f3, fps1, coorB, fqB,
                                                            N1, N2, 64, B);

  // stage 4: 128 queries vs 512 keys, edge-conv 128->256
  knn_kernel<<<dim3((N2 + 127) / 128, B), 128, 0, stream>>>(coorB, coorA, idx4, N2, N1);
  run_edge<64, 256, 128, 512>(fqB, f3, idx4, w4, g4, be4, stats, meanistd, f4, B, stream);

  // transposed outputs
  write_out<<<(B * N2 * 259 + 255) / 256, 256, 0, stream>>>(coorB, f4, out, B, N2, 256);
}